// Winograd_67516885893402
// MI455X (gfx1250) — compile-verified
//
#include <hip/hip_runtime.h>

typedef __attribute__((ext_vector_type(2))) float v2f;
typedef __attribute__((ext_vector_type(4))) float v4f;
typedef __attribute__((ext_vector_type(8))) float v8f;

// Only check for the builtin in the device pass; amdgcn builtins are not
// declared in the host (x86) pass of the HIP dual compilation.
#if defined(__HIP_DEVICE_COMPILE__) && !__has_builtin(__builtin_amdgcn_wmma_f32_16x16x4_f32)
#error "wmma_f32_16x16x4_f32 builtin not available on this toolchain"
#endif

// ---------------------------------------------------------------------------
// Compile-time construction of the fused Winograd operator:
//   vec(O)[4] = M[4x16] * vec(x)[16],   M[(i,l)][(p,q)] =
//     sum_{j,k} A[i,j] * A[l,k] * F'[j,k] * B[j,p] * B[k,q],  F' = G*FILT*G^T
// Padded to 16x16 (rows 4..15 zero) so it can be the WMMA A-matrix.
// ---------------------------------------------------------------------------
struct MMat { float v[16][16]; };

static constexpr MMat makeM() {
    const float Bm[4][4] = {{1,0,-1,0},{0,1,1,0},{0,-1,1,0},{0,1,0,-1}};
    const float G[4][3]  = {{1.0f,0.0f,0.0f},{0.5f,0.5f,0.5f},
                            {0.5f,-0.5f,0.5f},{0.0f,0.0f,1.0f}};
    const float Ft[3][3] = {{0,-1,0},{-1,4,-1},{0,-1,0}};
    const float A2[2][4] = {{1,1,1,0},{0,1,-1,-1}};

    float GF[4][3] = {};
    for (int i = 0; i < 4; ++i)
        for (int j = 0; j < 3; ++j) {
            float s = 0.f;
            for (int k = 0; k < 3; ++k) s += G[i][k] * Ft[k][j];
            GF[i][j] = s;
        }
    float Fp[4][4] = {};            // F' = G*FILT*G^T
    for (int i = 0; i < 4; ++i)
        for (int j = 0; j < 4; ++j) {
            float s = 0.f;
            for (int k = 0; k < 3; ++k) s += GF[i][k] * G[j][k];
            Fp[i][j] = s;
        }
    MMat m = {};                    // rows 4..15 stay zero
    for (int i = 0; i < 2; ++i)
        for (int l = 0; l < 2; ++l)
            for (int p = 0; p < 4; ++p)
                for (int q = 0; q < 4; ++q) {
                    float s = 0.f;
                    for (int j = 0; j < 4; ++j)
                        for (int k = 0; k < 4; ++k)
                            s += A2[i][j] * A2[l][k] * Fp[j][k] * Bm[j][p] * Bm[k][q];
                    m.v[i * 2 + l][p * 4 + q] = s;
                }
    return m;
}

__constant__ MMat cM = makeM();

// ---------------------------------------------------------------------------
// One wave handles 16 tiles per group iteration:
//   D(16x16) = Mpad(16x16, A-matrix) x X^T(16x16, B-matrix), chunked K=4 x4.
// D rows 0..3 = outputs 0..3, D cols = tile -> lanes 0..15 store one float4.
// ---------------------------------------------------------------------------
__global__ __launch_bounds__(256)
void winograd_wmma_kernel(const float* __restrict__ x,
                          float* __restrict__ out,
                          int ngroups) {
    const int lane = threadIdx.x & 31;
    const int mrow = lane & 15;              // A-matrix row / B-matrix column (tile)
    const int koff = ((lane >> 4) & 1) << 1; // lanes 16..31 hold K+2,K+3 of each chunk

    // Per-lane A fragments (constant Winograd operator), K chunks of 4.
    v2f a0, a1, a2, a3;
    a0.x = cM.v[mrow][0  + koff]; a0.y = cM.v[mrow][0  + koff + 1];
    a1.x = cM.v[mrow][4  + koff]; a1.y = cM.v[mrow][4  + koff + 1];
    a2.x = cM.v[mrow][8  + koff]; a2.y = cM.v[mrow][8  + koff + 1];
    a3.x = cM.v[mrow][12 + koff]; a3.y = cM.v[mrow][12 + koff + 1];

    const int wave   = (int)((blockIdx.x * blockDim.x + threadIdx.x) >> 5);
    const int nwaves = (int)((gridDim.x * blockDim.x) >> 5);

    for (int g = wave; g < ngroups; g += nwaves) {
        // B fragment: lane 'mrow' streams tile (g*16 + mrow), 2 floats per chunk.
        const float* tp = x + (((long)g * 16 + mrow) << 4) + koff;
        v2f b0 = *(const v2f*)(tp + 0);
        v2f b1 = *(const v2f*)(tp + 4);
        v2f b2 = *(const v2f*)(tp + 8);
        v2f b3 = *(const v2f*)(tp + 12);

        v8f c = {};
        c = __builtin_amdgcn_wmma_f32_16x16x4_f32(false, a0, false, b0, (short)0, c, false, false);
        c = __builtin_amdgcn_wmma_f32_16x16x4_f32(false, a1, false, b1, (short)0, c, false, false);
        c = __builtin_amdgcn_wmma_f32_16x16x4_f32(false, a2, false, b2, (short)0, c, false, false);
        c = __builtin_amdgcn_wmma_f32_16x16x4_f32(false, a3, false, b3, (short)0, c, false, false);

        // Lane n (<16) holds O[tile n][0..3] in c[0..3]: one coalesced b128 store.
        if (lane < 16) {
            v4f o;
            o.x = c[0]; o.y = c[1]; o.z = c[2]; o.w = c[3];
            *(v4f*)(out + (((long)g * 16 + lane) << 2)) = o;
        }
    }
}

extern "C" void kernel_launch(void* const* d_in, const int* in_sizes, int n_in,
                              void* d_out, int out_size, void* d_ws, size_t ws_size,
                              hipStream_t stream) {
    const float* x = (const float*)d_in[0];
    float* out = (float*)d_out;

    // in_sizes[0] = N_TILES * 16 elements; 16 tiles per wave-group.
    const int ntiles  = in_sizes[0] / 16;
    const int ngroups = ntiles / 16;

    const int threads = 256;                 // 8 wave32 per block
    int blocks = 4096;                       // 32768 waves; grid-stride over groups
    const int waves = blocks * (threads / 32);
    if (waves > ngroups) blocks = (ngroups + (threads / 32) - 1) / (threads / 32);
    if (blocks < 1) blocks = 1;

    winograd_wmma_kernel<<<blocks, threads, 0, stream>>>(x, out, ngroups);
}